// DySample_LP_57423712747899
// MI455X (gfx1250) — compile-verified
//
#include <hip/hip_runtime.h>

// DySample-LP: x(8,256,96,96) fp32 -> out(8,256,192,192) fp32
// Pass 1: 1x1-conv offset GEMM via V_WMMA_F32_16X16X4_F32 (M=73728,K=256,N=32)
// Pass 2: bilinear gather upsample (memory-bound; x stays resident in 192MB L2)

typedef __attribute__((ext_vector_type(2))) float v2f;
typedef __attribute__((ext_vector_type(8))) float v8f;

#define BB   8
#define CC   256
#define HH   96
#define WW   96
#define HWX  (HH * WW)          // 9216
#define OC   32
#define OH   192
#define OW   192
#define OHW  (OH * OW)          // 36864
#define CG   64                 // channels per group

// ---------------------------------------------------------------------------
// Kernel 1: offset[b, o, h, w] = (sum_c x[b,c,h,w] * w_off[o,c] + b_off[o]) * 0.25
// One wave handles a tile of 16 consecutive spatial positions (one M-tile),
// N=32 split into two 16-wide WMMA accumulators, K=256 in steps of 4.
// WMMA f32 16x16x4 layouts (ISA 7.12.2):
//   A (16x4):  lane L -> row M=L%16; VGPR0/1 hold K = (L/16)*2 + {0,1}
//   B (4x16):  lane L -> col N=L%16; VGPR0/1 hold K = (L/16)*2 + {0,1}
//   D (16x16): lane L -> col N=L%16; VGPR v holds row M = v + (L/16)*8
// ---------------------------------------------------------------------------
__global__ __launch_bounds__(256) void dysample_offset_gemm(
    const float* __restrict__ x, const float* __restrict__ w_off,
    const float* __restrict__ b_off, float* __restrict__ offs)
{
    const int lane = threadIdx.x & 31;
    const int wave = threadIdx.x >> 5;
    const int tile = blockIdx.x * 8 + wave;      // 4608 tiles total
    const int m0   = tile * 16;                  // first spatial index of tile
    const int b    = m0 / HWX;                   // tile never crosses batch
    const int hw0  = m0 - b * HWX;               // nor a W-row (W%16==0)

    const int mrow = lane & 15;                  // A row within tile
    const int kh   = lane >> 4;                  // K half (0 or 1)
    const int n    = lane & 15;                  // B/D column

    const float* xb = x + (size_t)b * CC * HWX + hw0 + mrow;
    const float* wb0 = w_off + (size_t)n * CC;
    const float* wb1 = w_off + (size_t)(n + 16) * CC;

    v8f acc0 = {};
    v8f acc1 = {};

#pragma unroll 4
    for (int k = 0; k < CC; k += 4) {
        const int kk = k + kh * 2;
        v2f a;
        a.x = xb[(size_t)kk * HWX];
        a.y = xb[(size_t)(kk + 1) * HWX];
        v2f bm0 = *(const v2f*)(wb0 + kk);        // 8B aligned: kk even
        v2f bm1 = *(const v2f*)(wb1 + kk);
        acc0 = __builtin_amdgcn_wmma_f32_16x16x4_f32(
            false, a, false, bm0, (short)0, acc0, false, false);
        acc1 = __builtin_amdgcn_wmma_f32_16x16x4_f32(
            false, a, false, bm1, (short)0, acc1, false, false);
    }

    const float bo0 = b_off[n];
    const float bo1 = b_off[n + 16];
    float* o0 = offs + ((size_t)b * OC + n) * HWX + hw0 + kh * 8;
    float* o1 = o0 + (size_t)16 * HWX;

#pragma unroll
    for (int v = 0; v < 8; ++v) {
        o0[v] = (acc0[v] + bo0) * 0.25f;
        o1[v] = (acc1[v] + bo1) * 0.25f;
    }
}

// ---------------------------------------------------------------------------
// Kernel 2: bilinear gather. One thread per (b, g, y, xo); coords computed
// once, then loop over the 64 channels of the group (coords are shared by
// all channels in a group). ix = clamp(w + offx, 0, W-1) after algebraic
// simplification of the grid normalization in the reference.
// ---------------------------------------------------------------------------
__global__ __launch_bounds__(256) void dysample_sample(
    const float* __restrict__ x, const float* __restrict__ offs,
    float* __restrict__ out)
{
    int idx = blockIdx.x * 256 + threadIdx.x;    // exact: 8*4*192*192 threads
    const int xo = idx % OW;
    int t = idx / OW;
    const int y = t % OH;
    t /= OH;
    const int g = t & 3;
    const int b = t >> 2;

    const int h = y >> 1, sy = y & 1;
    const int w = xo >> 1, sx = xo & 1;
    const int o = g * 4 + sy * 2 + sx;

    const size_t obase = ((size_t)b * OC + o) * HWX + h * WW + w;
    const float offx = offs[obase];
    const float offy = offs[obase + (size_t)16 * HWX];

    const float ix = fminf(fmaxf((float)w + offx, 0.0f), (float)(WW - 1));
    const float iy = fminf(fmaxf((float)h + offy, 0.0f), (float)(HH - 1));
    const float x0f = floorf(ix), y0f = floorf(iy);
    const float wx = ix - x0f,   wy = iy - y0f;
    const int x0 = (int)x0f,     y0 = (int)y0f;
    const int x1 = min(x0 + 1, WW - 1);
    const int y1 = min(y0 + 1, HH - 1);

    const float w00 = (1.0f - wy) * (1.0f - wx);
    const float w01 = (1.0f - wy) * wx;
    const float w10 = wy * (1.0f - wx);
    const float w11 = wy * wx;

    const int i00 = y0 * WW + x0, i01 = y0 * WW + x1;
    const int i10 = y1 * WW + x0, i11 = y1 * WW + x1;

    const float* xp = x   + ((size_t)b * CC + (size_t)g * CG) * HWX;
    float*       op = out + ((size_t)b * CC + (size_t)g * CG) * OHW
                          + (size_t)y * OW + xo;

#pragma unroll 4
    for (int c = 0; c < CG; ++c) {
        const float v00 = xp[i00];
        const float v01 = xp[i01];
        const float v10 = xp[i10];
        const float v11 = xp[i11];
        const float r = v00 * w00 + v01 * w01 + v10 * w10 + v11 * w11;
        __builtin_nontemporal_store(r, op);      // keep x resident in L2
        xp += HWX;
        op += OHW;
    }
}

// ---------------------------------------------------------------------------
extern "C" void kernel_launch(void* const* d_in, const int* in_sizes, int n_in,
                              void* d_out, int out_size, void* d_ws, size_t ws_size,
                              hipStream_t stream) {
    (void)in_sizes; (void)n_in; (void)out_size; (void)ws_size;
    const float* x     = (const float*)d_in[0];   // (8,256,96,96)
    const float* w_off = (const float*)d_in[1];   // (32,256)
    const float* b_off = (const float*)d_in[2];   // (32,)
    float* out  = (float*)d_out;                  // (8,256,192,192)
    float* offs = (float*)d_ws;                   // scratch: 8*32*96*96 f32 = 9.4 MB

    // Pass 1: 73728/16 = 4608 M-tiles, 8 waves (tiles) per 256-thread block
    dysample_offset_gemm<<<4608 / 8, 256, 0, stream>>>(x, w_off, b_off, offs);
    // Pass 2: 8*4*192*192 = 1179648 threads
    dysample_sample<<<(8 * 4 * OH * OW) / 256, 256, 0, stream>>>(x, offs, out);
}